// GATConv_21912923144576
// MI455X (gfx1250) — compile-verified
//
#include <hip/hip_runtime.h>
#include <hip/hip_bf16.h>

#define NNODES 50000
#define NEDGES 800000
#define IN_CHN 256
#define HEADS 4
#define OUTC 32
#define HC 128            // HEADS*OUTC
#define NEG_SLOPE 0.2f
#define EPSV 1e-16f

typedef __attribute__((ext_vector_type(2))) float v2f;
typedef __attribute__((ext_vector_type(8))) float v8f;

// order-preserving float<->uint mapping for atomicMax on floats
__device__ __forceinline__ unsigned fenc(float f) {
  unsigned u = __float_as_uint(f);
  return (u & 0x80000000u) ? ~u : (u | 0x80000000u);
}
__device__ __forceinline__ float fdec(unsigned e) {
  return (e & 0x80000000u) ? __uint_as_float(e ^ 0x80000000u) : __uint_as_float(~e);
}

// ---------------------------------------------------------------------------
// K1: h = x @ W   (x:[N,256], W view:[256,128] with W[k][j]=weight[j>>5][k][j&31])
// One wave -> 16 rows x 128 cols, K=256 via 64 x v_wmma_f32_16x16x4_f32 x 8 tiles
// ---------------------------------------------------------------------------
__global__ void k_gemm(const float* __restrict__ x, const float* __restrict__ w,
                       float* __restrict__ h) {
  const int wave = (int)((blockIdx.x * blockDim.x + threadIdx.x) >> 5);
  const int lane = (int)(threadIdx.x & 31u);
  if (wave >= NNODES / 16) return;            // wave-uniform exit; EXEC stays all-1s
  const int r    = lane & 15;                 // row within tile (A), col within tile (B/C)
  const int half = lane >> 4;                 // half-wave selects K pair / row block

  const float* xrow = x + (size_t)(wave * 16 + r) * IN_CHN + 2 * half;

  const float* bptr[8];
#pragma unroll
  for (int t = 0; t < 8; ++t) {
    int col = t * 16 + r;                     // global output column 0..127
    bptr[t] = w + (size_t)(col >> 5) * (IN_CHN * OUTC) + (col & 31);
  }

  const v8f vzero = {0.f, 0.f, 0.f, 0.f, 0.f, 0.f, 0.f, 0.f};
  v8f acc[8] = {vzero, vzero, vzero, vzero, vzero, vzero, vzero, vzero};

  for (int k = 0; k < IN_CHN; k += 4) {
    v2f a = *(const v2f*)(xrow + k);          // x[row][k + 2*half + {0,1}]
    const int kb = (k + 2 * half) * OUTC;     // B element row stride = 32 floats
#pragma unroll
    for (int t = 0; t < 8; ++t) {
      v2f b;
      b.x = bptr[t][kb];
      b.y = bptr[t][kb + OUTC];
      acc[t] = __builtin_amdgcn_wmma_f32_16x16x4_f32(
          false, a, false, b, (short)0, acc[t], false, false);
    }
  }

  // C layout: VGPR v, lane -> row (v + 8*half), col r   (coalesced b32 stores)
  float* hp = h + (size_t)(wave * 16 + 8 * half) * HC + r;
#pragma unroll
  for (int t = 0; t < 8; ++t)
#pragma unroll
    for (int v = 0; v < 8; ++v)
      hp[(size_t)v * HC + t * 16] = acc[t][v];
}

// ---------------------------------------------------------------------------
// K2: per-(node,head) attention logits  a_src/a_dst = <h[n,hd,:], att[hd,:32/32:]>
// ---------------------------------------------------------------------------
__global__ void k_att(const float* __restrict__ h, const float* __restrict__ att,
                      float* __restrict__ asrc, float* __restrict__ adst) {
  int i = (int)(blockIdx.x * blockDim.x + threadIdx.x);
  if (i >= NNODES * HEADS) return;
  int n = i >> 2, hd = i & 3;
  const float4* hp = (const float4*)(h + (size_t)n * HC + hd * OUTC);
  const float4* as = (const float4*)(att + (size_t)hd * 2 * OUTC);
  const float4* ad = as + OUTC / 4;
  float s = 0.f, d = 0.f;
#pragma unroll
  for (int q = 0; q < OUTC / 4; ++q) {
    float4 hv = hp[q], av = as[q], bv = ad[q];
    s += hv.x * av.x + hv.y * av.y + hv.z * av.z + hv.w * av.w;
    d += hv.x * bv.x + hv.y * bv.y + hv.z * bv.z + hv.w * bv.w;
  }
  asrc[i] = s;
  adst[i] = d;
}

// ---------------------------------------------------------------------------
// K0: init accumulators (out=0, amax=enc(-1e9), sum_exp=0)
// ---------------------------------------------------------------------------
__global__ void k_init(float* __restrict__ out, unsigned* __restrict__ amax,
                       float* __restrict__ sume) {
  int i = (int)(blockIdx.x * blockDim.x + threadIdx.x);
  if (i >= NNODES * HC) return;
  out[i] = 0.f;
  int k = i & (HC - 1);
  if (k < HEADS) {
    int n = i >> 7;
    amax[n * HEADS + k] = fenc(-1e9f);
    sume[n * HEADS + k] = 0.f;
  }
}

// ---------------------------------------------------------------------------
// K3: segment max of leaky-relu logits over incoming edges
// ---------------------------------------------------------------------------
__global__ void k_logit_max(const int* __restrict__ src, const int* __restrict__ dst,
                            const float* __restrict__ asrc, const float* __restrict__ adst,
                            unsigned* __restrict__ amax) {
  int i = (int)(blockIdx.x * blockDim.x + threadIdx.x);
  if (i >= NEDGES * HEADS) return;
  int e = i >> 2, hd = i & 3;
  int s = src[e], d = dst[e];
  float al = asrc[s * HEADS + hd] + adst[d * HEADS + hd];
  al = al > 0.f ? al : NEG_SLOPE * al;
  atomicMax(&amax[d * HEADS + hd], fenc(al));
}

// ---------------------------------------------------------------------------
// K4: sum of exp(alpha - max) per (dst,head)
// ---------------------------------------------------------------------------
__global__ void k_logit_sum(const int* __restrict__ src, const int* __restrict__ dst,
                            const float* __restrict__ asrc, const float* __restrict__ adst,
                            const unsigned* __restrict__ amax, float* __restrict__ sume) {
  int i = (int)(blockIdx.x * blockDim.x + threadIdx.x);
  if (i >= NEDGES * HEADS) return;
  int e = i >> 2, hd = i & 3;
  int s = src[e], d = dst[e];
  float al = asrc[s * HEADS + hd] + adst[d * HEADS + hd];
  al = al > 0.f ? al : NEG_SLOPE * al;
  float p = __expf(al - fdec(amax[d * HEADS + hd]));
  atomicAdd(&sume[d * HEADS + hd], p);
}

// ---------------------------------------------------------------------------
// K5: unnormalized weighted scatter-add: out[dst] += exp(alpha-max) * h[src]
// 128 consecutive lanes per edge -> coalesced 512B gather + 512B atomic burst
// ---------------------------------------------------------------------------
__global__ void k_scatter(const int* __restrict__ src, const int* __restrict__ dst,
                          const float* __restrict__ asrc, const float* __restrict__ adst,
                          const unsigned* __restrict__ amax,
                          const float* __restrict__ h, float* __restrict__ out) {
  int e = (int)(blockIdx.x * 2 + (threadIdx.x >> 7));
  if (e >= NEDGES) return;
  int k  = (int)(threadIdx.x & 127u);
  int hd = k >> 5;
  int s = src[e], d = dst[e];
  float al = asrc[s * HEADS + hd] + adst[d * HEADS + hd];
  al = al > 0.f ? al : NEG_SLOPE * al;
  float p = __expf(al - fdec(amax[d * HEADS + hd]));
  atomicAdd(&out[(size_t)d * HC + k], p * h[(size_t)s * HC + k]);
}

// ---------------------------------------------------------------------------
// K6: deferred normalization + bias:  out = out/(sum+eps) + bias
// ---------------------------------------------------------------------------
__global__ void k_final(float* __restrict__ out, const float* __restrict__ sume,
                        const float* __restrict__ bias) {
  int i = (int)(blockIdx.x * blockDim.x + threadIdx.x);
  if (i >= NNODES * HC) return;
  int n = i >> 7, k = i & 127, hd = k >> 5;
  out[i] = out[i] / (sume[n * HEADS + hd] + EPSV) + bias[k];
}

extern "C" void kernel_launch(void* const* d_in, const int* in_sizes, int n_in,
                              void* d_out, int out_size, void* d_ws, size_t ws_size,
                              hipStream_t stream) {
  const float* x    = (const float*)d_in[0];
  const int*   ei   = (const int*)d_in[1];
  const float* w    = (const float*)d_in[2];
  const float* att  = (const float*)d_in[3];
  const float* bias = (const float*)d_in[4];
  float* out = (float*)d_out;

  // workspace layout: h[N*128] | a_src[N*4] | a_dst[N*4] | amax[N*4] | sum[N*4]
  float*    h    = (float*)d_ws;
  float*    asrc = h + (size_t)NNODES * HC;
  float*    adst = asrc + (size_t)NNODES * HEADS;
  unsigned* amax = (unsigned*)(adst + (size_t)NNODES * HEADS);
  float*    sume = (float*)(amax + (size_t)NNODES * HEADS);

  const int* src = ei;
  const int* dst = ei + NEDGES;

  k_init<<<(NNODES * HC + 255) / 256, 256, 0, stream>>>(out, amax, sume);
  k_gemm<<<(NNODES / 16 + 3) / 4, 128, 0, stream>>>(x, w, h);
  k_att<<<(NNODES * HEADS + 255) / 256, 256, 0, stream>>>(h, att, asrc, adst);
  k_logit_max<<<(NEDGES * HEADS + 255) / 256, 256, 0, stream>>>(src, dst, asrc, adst, amax);
  k_logit_sum<<<(NEDGES * HEADS + 255) / 256, 256, 0, stream>>>(src, dst, asrc, adst, amax, sume);
  k_scatter<<<NEDGES / 2, 256, 0, stream>>>(src, dst, asrc, adst, amax, h, out);
  k_final<<<(NNODES * HC + 255) / 256, 256, 0, stream>>>(out, sume, bias);
}